// SDPAChunkedKV_49478023250147
// MI455X (gfx1250) — compile-verified
//
#include <hip/hip_runtime.h>
#include <hip/hip_bf16.h>
#include <math.h>

#define HIDDEN 4096
#define HEADS 32
#define HDIM 128
#define KV_LEN 8192
#define BATCH 8

typedef float v2f __attribute__((ext_vector_type(2)));
typedef float v8f __attribute__((ext_vector_type(8)));
typedef unsigned int v4u __attribute__((ext_vector_type(4)));
typedef int v4i __attribute__((ext_vector_type(4)));
typedef int v8i __attribute__((ext_vector_type(8)));

union V8 { v8f v; float f[8]; };

__device__ __forceinline__ unsigned rfl(unsigned x) {
  return (unsigned)__builtin_amdgcn_readfirstlane((int)x);
}

// -----------------------------------------------------------------------------
// TDM: DMA one 2D tile (tile_d rows x 16 kv, f32) from global into LDS.
// D# per CDNA5 ISA 8.3/8.4: group0 = {ctrl, lds_addr, gaddr_lo, gaddr_hi|type},
// group1 packs data_size / tensor dims / tile dims / row stride.
// Tracked on TENSORcnt; caller pipelines with s_wait_tensorcnt.
// -----------------------------------------------------------------------------
__device__ __forceinline__ void tdm_load_k_chunk(unsigned lds_addr,
                                                 unsigned long long ga) {
  v4u g0 = { 1u,                                        // count=1 (valid), user
             lds_addr,                                  // LDS byte address
             (unsigned)ga,                              // global addr [31:0]
             ((unsigned)(ga >> 32) & 0x01FFFFFFu)       // global addr [56:32]
                 | 0x80000000u };                       // type=2 ("image")
  v8i g1 = { 0x20000,                    // data_size=4B, no multicast/flags
             (int)(((unsigned)KV_LEN & 0xFFFFu) << 16), // tensor_dim0 lo16
             (int)(128u << 16),          // tensor_dim0 hi=0 | tensor_dim1=128
             (int)(16u << 16),           // tensor_dim1 hi=0 | tile_dim0=16 (kv)
             16,                         // tile_dim1=16 (d rows), tile_dim2=0
             (int)KV_LEN,                // tensor_dim0_stride[31:0] (elements)
             0, 0 };                     // stride hi / dim1_stride unused (2D)
  v4i gz = { 0, 0, 0, 0 };
#if defined(__clang_major__) && __clang_major__ >= 23
  v8i gz8 = { 0, 0, 0, 0, 0, 0, 0, 0 };
  __builtin_amdgcn_tensor_load_to_lds(g0, g1, gz, gz, gz8, 0);
#else
  __builtin_amdgcn_tensor_load_to_lds(g0, g1, gz, gz, 0);
#endif
}

// -----------------------------------------------------------------------------
// out[M=BATCH][N] = act[M][K] @ W[N][K]^T  using V_WMMA_F32_16X16X4_F32.
// One wave per 16-column tile.  Lanes with tile-row m>=8 load duplicates of
// rows m&7 (same addresses -> coalesced, no EXEC churn); their D rows are
// simply never written back.
// -----------------------------------------------------------------------------
__global__ void proj_wmma_kernel(const float* __restrict__ act,
                                 const float* __restrict__ W,
                                 float* __restrict__ out,
                                 int N, int K) {
  const int lane = threadIdx.x & 31;
  const int wave = threadIdx.x >> 5;
  const int tile = blockIdx.x * (blockDim.x >> 5) + wave;
  const long n0  = (long)tile * 16;

  const int kq = (lane >> 4) << 1;    // 0 or 2: k-pair owned by this half-wave
  const float* wrow = W + (n0 + (lane & 15)) * (long)K + kq;
  const float* arow = act + (long)(lane & 7) * K + kq;   // rows 8-15 duplicate

  v8f acc = {};
#pragma unroll 4
  for (int k0 = 0; k0 < K; k0 += 4) {
    const float2 av = *(const float2*)(arow + k0);
    const float2 bv = *(const float2*)(wrow + k0);
    v2f a, bf;
    a.x = av.x;  a.y = av.y;
    bf.x = bv.x; bf.y = bv.y;
    acc = __builtin_amdgcn_wmma_f32_16x16x4_f32(false, a, false, bf,
                                                (short)0, acc, false, false);
  }

  // D layout: VGPR j, lanes 0-15 -> row M=j at column N=lane.
  if (lane < 16) {
    V8 u; u.v = acc;
#pragma unroll
    for (int j = 0; j < BATCH; ++j)
      out[(long)j * N + n0 + lane] = u.f[j];
  }
}

// -----------------------------------------------------------------------------
// Fused single-token attention, one (b,h) per workgroup (256 thr = 8 waves).
//   Phase 1: scores via f32 WMMA; K tiles streamed by the Tensor Data Mover
//            into double-buffered LDS (s_wait_tensorcnt pipeline).
//   Phase 2: block softmax over the 8192 scores staged in LDS.
//   Phase 3: o[d] = sum_l p[l] * V[d][l] — contiguous-l VALU float4 dot.
// -----------------------------------------------------------------------------
__global__ void attn_decode_kernel(const float* __restrict__ q,    // [B, H*D]
                                   const float* __restrict__ kc,   // [B, H*D, L]
                                   const float* __restrict__ vc,   // [B, H*D, L]
                                   float* __restrict__ attn) {     // [B, H*D]
  __shared__ __align__(16) float sc[KV_LEN];          // scores -> probs (32KB)
  __shared__ __align__(16) float kbuf[8][2][16 * 16]; // per-wave dbl buf (16KB)
  __shared__ float qs[HDIM];
  __shared__ float red[256];

  const int bh   = blockIdx.x;
  const int b    = bh / HEADS;
  const int h    = bh % HEADS;
  const int tid  = threadIdx.x;
  const int lane = tid & 31;
  const int wave = tid >> 5;

  const long base = ((long)b * (HEADS * HDIM) + (long)h * HDIM) * KV_LEN;
  const float* Kb = kc + base;
  const float* Vb = vc + base;
  const float scale = 0.08838834764831845f;           // 1/sqrt(128)

  if (tid < HDIM)
    qs[tid] = q[(long)b * (HEADS * HDIM) + h * HDIM + tid] * scale;
  __syncthreads();

  // ---- Phase 1: scores via TDM-fed V_WMMA_F32_16X16X4_F32 -----------------
  const int lrow = lane & 15;
  const int dq   = (lane >> 4) << 1;
  const unsigned ldsA =
      rfl((unsigned)(uintptr_t)&kbuf[wave][0][0]);    // LDS byte offsets
  const unsigned ldsB =
      rfl((unsigned)(uintptr_t)&kbuf[wave][1][0]);
  const unsigned long long kga = (unsigned long long)(uintptr_t)Kb;

  for (int t = wave; t < KV_LEN / 16; t += 8) {
    const unsigned off0 = rfl((unsigned)(t * 16 * 4));     // l0 byte offset
    // chunk c = d rows [16c, 16c+16); prime the pipeline with chunk 0.
    tdm_load_k_chunk(ldsA, kga + off0);
    v8f acc = {};
#pragma unroll
    for (int c = 0; c < 8; ++c) {
      if (c < 7) {
        // issue chunk c+1 into the other buffer while consuming chunk c
        const unsigned coff = (unsigned)((c + 1) * 16 * KV_LEN * 4);
        tdm_load_k_chunk((c & 1) ? ldsA : ldsB, kga + off0 + coff);
        __builtin_amdgcn_s_wait_tensorcnt((short)1);  // chunk c landed
      } else {
        __builtin_amdgcn_s_wait_tensorcnt((short)0);  // drain
      }
      const float* lb = &kbuf[wave][c & 1][0];        // [d][l] rows of 16
#pragma unroll
      for (int dd = 0; dd < 16; dd += 4) {
        v2f a, bf;
        a.x  = lb[(dd + dq) * 16 + lrow];
        a.y  = lb[(dd + dq + 1) * 16 + lrow];
        bf.x = qs[c * 16 + dd + dq];
        bf.y = qs[c * 16 + dd + dq + 1];
        acc = __builtin_amdgcn_wmma_f32_16x16x4_f32(false, a, false, bf,
                                                    (short)0, acc, false,
                                                    false);
      }
    }
    // Column N=0 of D: lane 0 holds rows 0..7, lane 16 holds rows 8..15.
    V8 u; u.v = acc;
    const int l0 = t * 16;
    if (lane == 0) {
#pragma unroll
      for (int j = 0; j < 8; ++j) sc[l0 + j] = u.f[j];
    } else if (lane == 16) {
#pragma unroll
      for (int j = 0; j < 8; ++j) sc[l0 + 8 + j] = u.f[j];
    }
  }
  __syncthreads();

  // ---- Phase 2: softmax over 8192 scores ----------------------------------
  float lmax = -INFINITY;
  for (int i = tid; i < KV_LEN; i += 256) lmax = fmaxf(lmax, sc[i]);
  red[tid] = lmax;
  __syncthreads();
  for (int s = 128; s > 0; s >>= 1) {
    if (tid < s) red[tid] = fmaxf(red[tid], red[tid + s]);
    __syncthreads();
  }
  const float gmax = red[0];
  __syncthreads();

  float lsum = 0.0f;
  for (int i = tid; i < KV_LEN; i += 256) {
    const float p = __expf(sc[i] - gmax);
    sc[i] = p;
    lsum += p;
  }
  red[tid] = lsum;
  __syncthreads();
  for (int s = 128; s > 0; s >>= 1) {
    if (tid < s) red[tid] += red[tid + s];
    __syncthreads();
  }
  const float inv = 1.0f / red[0];
  __syncthreads();

  // ---- Phase 3: o[d] = sum_l p[l] * V[d][l]  (contiguous-l VALU dot) ------
  const int d0 = wave * 16;                           // 8 waves x 16 d = 128
#pragma unroll 1
  for (int dd = 0; dd < 16; ++dd) {
    const float* vrow = Vb + (long)(d0 + dd) * KV_LEN;
    float acc = 0.0f;
    for (int l = lane * 4; l < KV_LEN; l += 128) {
      __builtin_prefetch(vrow + l + 512, 0, 1);
      const float4 vv = *(const float4*)(vrow + l);
      const float4 pp = *(const float4*)(sc + l);
      acc += vv.x * pp.x + vv.y * pp.y + vv.z * pp.z + vv.w * pp.w;
    }
#pragma unroll
    for (int off = 16; off > 0; off >>= 1) acc += __shfl_xor(acc, off, 32);
    if (lane == 0)
      attn[(long)b * (HEADS * HDIM) + h * HDIM + d0 + dd] = acc * inv;
  }
}

// -----------------------------------------------------------------------------
extern "C" void kernel_launch(void* const* d_in, const int* in_sizes, int n_in,
                              void* d_out, int out_size, void* d_ws, size_t ws_size,
                              hipStream_t stream) {
  const float* x  = (const float*)d_in[0];
  const float* kc = (const float*)d_in[1];
  const float* vc = (const float*)d_in[2];
  const float* Wq = (const float*)d_in[3];
  const float* Wo = (const float*)d_in[4];
  float* out    = (float*)d_out;
  float* qws    = (float*)d_ws;                     // [8, 4096]
  float* attnws = qws + (size_t)BATCH * HIDDEN;     // [8, 4096]

  // 256 column tiles, 4 waves (128 thr) per block -> 64 blocks.
  proj_wmma_kernel<<<dim3((HIDDEN / 16) / 4), dim3(128), 0, stream>>>(
      x, Wq, qws, HIDDEN, HIDDEN);

  attn_decode_kernel<<<dim3(BATCH * HEADS), dim3(256), 0, stream>>>(
      qws, kc, vc, attnws);

  proj_wmma_kernel<<<dim3((HIDDEN / 16) / 4), dim3(128), 0, stream>>>(
      attnws, Wo, out, HIDDEN, HIDDEN);
}